// FeatureTransformer_5909875000395
// MI455X (gfx1250) — compile-verified
//
#include <hip/hip_runtime.h>
#include <stdint.h>

// ---------------------------------------------------------------------------
// NNUE feature-transformer for MI455X (gfx1250).
//
// The workload is a sparse row-gather + sum (no dense matmul), so the CDNA5
// path that matters is the Tensor Data Mover in GATHER mode: one
// TENSOR_LOAD_TO_LDS descriptor DMAs 16 table rows (16-bit row indices) into
// LDS asynchronously (TENSORcnt). Two descriptors cover the 32 active
// features of one batch row; we pipeline wait(1)/process/wait(0)/process.
// ---------------------------------------------------------------------------

typedef unsigned int v4u  __attribute__((ext_vector_type(4)));
typedef int          v8i_ __attribute__((ext_vector_type(8)));
typedef int          v4i_ __attribute__((ext_vector_type(4)));

#define FT_N_IN      41024
#define FT_N_OUT     256
#define FT_ROW_ELEMS 257      // N_OUT + 1 (psqt column)
#define FT_MAX_ACT   32

__global__ __launch_bounds__(32)
void ft_gather_tdm_kernel(const int*   __restrict__ ft_ics,   // [batch, 32]
                          const float* __restrict__ weight,   // [41024, 257]
                          const float* __restrict__ bias,     // [256]
                          float*       __restrict__ out_fts,  // [batch, 256]
                          float*       __restrict__ out_psqt, // [batch]
                          int batch)
{
    __shared__ float smem[FT_MAX_ACT * FT_ROW_ELEMS];   // 32 rows * 1028 B = 32896 B

    const int row  = blockIdx.x;
    if (row >= batch) return;
    const int lane = threadIdx.x;           // 0..31, one wave32 per workgroup

    // ---- load this row's 32 feature indices (one per lane) ----------------
    int idx = ft_ics[(size_t)row * FT_MAX_ACT + lane];
    unsigned mask = __builtin_amdgcn_ballot_w32(idx >= 0);  // wave-uniform
    int sidx = idx < 0 ? 0 : idx;           // clamp pads to a valid row; masked later

    // pack 16-bit row indices: p[i] = idx(2i) | idx(2i+1)<<16
    unsigned p[16];
#pragma unroll
    for (int i = 0; i < 16; ++i) {
        unsigned lo = (unsigned)__builtin_amdgcn_readlane(sidx, 2 * i);
        unsigned hi = (unsigned)__builtin_amdgcn_readlane(sidx, 2 * i + 1);
        p[i] = (lo & 0xFFFFu) | ((hi & 0xFFFFu) << 16);
    }

    // ---- build Tensor DMA descriptors (D#, ISA cdna5 ch.8) ----------------
    const uint64_t gaddr = (uint64_t)(uintptr_t)weight;
    const unsigned lds0  = (unsigned)(uintptr_t)(&smem[0]); // low 32 bits = LDS offset

    // Group 0: count=1 | gather_index_size=0 (16-bit) | gather_mode=1 ; lds ; gaddr ; type=2
    v4u g0;
    g0[0] = 0x80000001u;
    g0[1] = lds0;
    g0[2] = (unsigned)gaddr;
    g0[3] = ((unsigned)(gaddr >> 32) & 0x01FFFFFFu) | (2u << 30);

    // Group 1: data_size=4B; tensor_dim0=257; tensor_dim1=41024; tile_dim0=257;
    //          tile_dim1=16 (#indices); tensor_dim0_stride=257
    v8i_ g1;
    g1[0] = (int)(2u << 16);                                    // data_size = 2 (4 bytes)
    g1[1] = (int)(((unsigned)FT_ROW_ELEMS & 0xFFFFu) << 16);    // tensor_dim0[15:0]
    g1[2] = (int)(((unsigned)FT_N_IN & 0xFFFFu) << 16);         // dim0[31:16]=0 | dim1[15:0]
    g1[3] = (int)(((unsigned)FT_ROW_ELEMS) << 16);              // dim1[31:16]=0 | tile_dim0
    g1[4] = 16;                                                 // tile_dim1 = #valid indices
    g1[5] = FT_ROW_ELEMS;                                       // tensor_dim0_stride[31:0]
    g1[6] = 0;                                                  // stride[47:32]=0
    g1[7] = 0;

    // Groups 2/3: 16-bit row indices (8 per group); also provide the same
    // 8 index dwords in the combined int32x8 slot of the 6-arg builtin.
    v4i_ g2a = { (int)p[0],  (int)p[1],  (int)p[2],  (int)p[3]  };
    v4i_ g3a = { (int)p[4],  (int)p[5],  (int)p[6],  (int)p[7]  };
    v8i_ gia = { (int)p[0],  (int)p[1],  (int)p[2],  (int)p[3],
                 (int)p[4],  (int)p[5],  (int)p[6],  (int)p[7]  };
    v4i_ g2b = { (int)p[8],  (int)p[9],  (int)p[10], (int)p[11] };
    v4i_ g3b = { (int)p[12], (int)p[13], (int)p[14], (int)p[15] };
    v8i_ gib = { (int)p[8],  (int)p[9],  (int)p[10], (int)p[11],
                 (int)p[12], (int)p[13], (int)p[14], (int)p[15] };

    // ---- issue two async TDM gathers (rows 0-15, rows 16-31) --------------
    __builtin_amdgcn_tensor_load_to_lds(g0, g1, g2a, g3a, gia, 0);
    v4u g0b = g0;
    g0b[1] = lds0 + 16u * FT_ROW_ELEMS * 4u;
    __builtin_amdgcn_tensor_load_to_lds(g0b, g1, g2b, g3b, gib, 0);

    // ---- masked accumulate, pipelined against the second DMA --------------
    float acc[8] = {0.f, 0.f, 0.f, 0.f, 0.f, 0.f, 0.f, 0.f};
    float accP   = 0.f;

    __builtin_amdgcn_s_wait_tensorcnt(1);   // first descriptor complete
#pragma unroll 4
    for (int k = 0; k < 16; ++k) {
        if (mask & (1u << k)) {             // wave-uniform branch: skip padded rows
            const float* r = smem + k * FT_ROW_ELEMS;
#pragma unroll
            for (int j = 0; j < 8; ++j) acc[j] += r[lane + 32 * j];
            accP += r[FT_N_OUT];            // LDS broadcast
        }
    }

    __builtin_amdgcn_s_wait_tensorcnt(0);   // second descriptor complete
#pragma unroll 4
    for (int k = 16; k < 32; ++k) {
        if (mask & (1u << k)) {
            const float* r = smem + k * FT_ROW_ELEMS;
#pragma unroll
            for (int j = 0; j < 8; ++j) acc[j] += r[lane + 32 * j];
            accP += r[FT_N_OUT];
        }
    }

    // ---- bias add (bias_full = [0, bias]) and coalesced stores ------------
    float* fts = out_fts + (size_t)row * FT_N_OUT;
#pragma unroll
    for (int j = 0; j < 8; ++j) {
        int col  = lane + 32 * j;
        float bv = (col == 0) ? 0.0f : bias[col - 1];
        fts[col] = acc[j] + bv;
    }
    if (lane == 0) out_psqt[row] = accP + bias[FT_N_OUT - 1];
}

extern "C" void kernel_launch(void* const* d_in, const int* in_sizes, int n_in,
                              void* d_out, int out_size, void* d_ws, size_t ws_size,
                              hipStream_t stream)
{
    (void)n_in; (void)d_ws; (void)ws_size; (void)out_size;

    const int*   ft_ics = (const int*)  d_in[0];   // [batch, 32] int32
    const float* weight = (const float*)d_in[1];   // [41024, 257] fp32
    const float* bias   = (const float*)d_in[2];   // [256] fp32

    const int batch = in_sizes[0] / FT_MAX_ACT;    // 16384

    float* out      = (float*)d_out;
    float* out_fts  = out;                                   // batch*256
    float* out_psqt = out + (size_t)batch * FT_N_OUT;        // batch*1

    ft_gather_tdm_kernel<<<batch, 32, 0, stream>>>(ft_ics, weight, bias,
                                                   out_fts, out_psqt, batch);
}